// ReLUMaxPool2dKAN_25434796327098
// MI455X (gfx1250) — compile-verified
//
#include <hip/hip_runtime.h>

// ReLU-MaxPool 2x2 stride 2 over (16, 64, 224, 224) fp32 -> (16, 64, 112, 112) fp32.
// Pure streaming kernel: ~257 MB moved once, ~11 us floor at 23.3 TB/s HBM.
// B128 non-temporal loads/stores, 4 outputs per thread, exact reference max-tree order.

typedef float f4 __attribute__((ext_vector_type(4)));

#define POOL_H   224
#define POOL_W   224
#define POOL_HO  112
#define POOL_WO  112
#define POOL_NCH (16 * 64)                        // B*C fused channels
#define POOL_ROWS (POOL_NCH * POOL_HO)            // 114688 output rows
#define POOL_V4_PER_ROW (POOL_WO / 4)             // 28 float4 stores per output row
#define POOL_TOTAL_V4 (POOL_ROWS * POOL_V4_PER_ROW) // 3,211,264 (divisible by 256)
#define POOL_BLOCK 256

// Faithful mirror of the reference: max(a,b) = b + relu(a-b)
__device__ __forceinline__ float relu_max(float a, float b) {
    return b + fmaxf(a - b, 0.0f);
}

__global__ __launch_bounds__(POOL_BLOCK)
void relu_maxpool2x2_kernel(const float* __restrict__ in, float* __restrict__ out) {
    // One thread = 4 output columns = one float4 store.
    unsigned v   = blockIdx.x * POOL_BLOCK + threadIdx.x;   // [0, POOL_TOTAL_V4)
    unsigned row = v / POOL_V4_PER_ROW;                     // fused (b,c,oy) output row
    unsigned c4  = v % POOL_V4_PER_ROW;                     // float4 index within row
    unsigned ch  = row / POOL_HO;                           // fused (b,c) channel
    unsigned oy  = row % POOL_HO;                           // output y

    // Input base: 8 consecutive floats of input row y0 = 2*oy (16B-aligned).
    unsigned in_off = ch * (POOL_H * POOL_W) + (oy * 2) * POOL_W + c4 * 8;
    const f4* p_top = reinterpret_cast<const f4*>(in + in_off);
    const f4* p_bot = reinterpret_cast<const f4*>(in + in_off + POOL_W);

    // Streamed exactly once -> non-temporal (TH_NT) so we don't thrash the 192MB L2.
    f4 t0 = __builtin_nontemporal_load(p_top);       // top row, cols x..x+3
    f4 t1 = __builtin_nontemporal_load(p_top + 1);   // top row, cols x+4..x+7
    f4 b0 = __builtin_nontemporal_load(p_bot);       // bottom row, cols x..x+3
    f4 b1 = __builtin_nontemporal_load(p_bot + 1);   // bottom row, cols x+4..x+7

    // Reference tree per window [p00, p01, p10, p11]:
    //   m_top = rm(p00, p01); m_bot = rm(p10, p11); out = rm(m_top, m_bot)
    f4 o;
    {
        float mt = relu_max(t0.x, t0.y);
        float mb = relu_max(b0.x, b0.y);
        o.x = relu_max(mt, mb);
    }
    {
        float mt = relu_max(t0.z, t0.w);
        float mb = relu_max(b0.z, b0.w);
        o.y = relu_max(mt, mb);
    }
    {
        float mt = relu_max(t1.x, t1.y);
        float mb = relu_max(b1.x, b1.y);
        o.z = relu_max(mt, mb);
    }
    {
        float mt = relu_max(t1.z, t1.w);
        float mb = relu_max(b1.z, b1.w);
        o.w = relu_max(mt, mb);
    }

    f4* op = reinterpret_cast<f4*>(out + row * POOL_WO + c4 * 4);
    __builtin_nontemporal_store(o, op);
}

extern "C" void kernel_launch(void* const* d_in, const int* in_sizes, int n_in,
                              void* d_out, int out_size, void* d_ws, size_t ws_size,
                              hipStream_t stream) {
    (void)in_sizes; (void)n_in; (void)out_size; (void)d_ws; (void)ws_size;
    const float* x = reinterpret_cast<const float*>(d_in[0]);
    float* out = reinterpret_cast<float*>(d_out);

    const unsigned grid = POOL_TOTAL_V4 / POOL_BLOCK;  // 12544 blocks of 8 wave32s
    relu_maxpool2x2_kernel<<<grid, POOL_BLOCK, 0, stream>>>(x, out);
}